// DiscreteContinuousConvS2_48378511622745
// MI455X (gfx1250) — compile-verified
//
#include <hip/hip_runtime.h>

// Problem constants (fixed by the reference)
#define HIN   181
#define WIN   360
#define HOUT  91
#define WOUT  180
#define BATCH 4
#define CIN   32
#define COUT  32
#define KSIZE 9              // (3-1)*4+1
#define BC    (BATCH*CIN)    // 128
#define CK    (CIN*KSIZE)    // 288
#define HWIN  (HIN*WIN)      // 65160
#define HWOUT (HOUT*WOUT)    // 16380
#define NPAD  16384          // padded N for even 16-wide tiling
#define NSEG  (KSIZE*HOUT)   // 819
#define PSCALE 2

typedef float v2f __attribute__((ext_vector_type(2)));
typedef float v8f __attribute__((ext_vector_type(8)));

// ---------------- stage 0a: zero padded y ----------------
__global__ void k_zero(float* __restrict__ p, int n) {
    int i = blockIdx.x * blockDim.x + threadIdx.x;
    int stride = gridDim.x * blockDim.x;
    for (; i < n; i += stride) p[i] = 0.0f;
}

// ---------------- stage 0b: segment run bounds ----------------
__global__ void k_seg_init(int* __restrict__ segStart, int* __restrict__ segEnd) {
    int s = blockIdx.x * blockDim.x + threadIdx.x;
    if (s < NSEG) { segStart[s] = 0; segEnd[s] = 0; }
}

// psi_idx rows: [0]=k, [1]=t, [2]=flat.  Entries are sorted by t then k,
// so each segment s = k*HOUT + t is a single contiguous run (possibly empty).
__global__ void k_seg_mark(const long long* __restrict__ idx, int nnz,
                           int* __restrict__ segStart, int* __restrict__ segEnd) {
    int e = blockIdx.x * blockDim.x + threadIdx.x;
    if (e >= nnz) return;
    const long long* kk = idx;
    const long long* tt = idx + (size_t)nnz;
    int s = (int)(kk[e] * HOUT + tt[e]);
    bool first = (e == 0)       || ((int)(kk[e-1] * HOUT + tt[e-1]) != s);
    bool last  = (e == nnz - 1) || ((int)(kk[e+1] * HOUT + tt[e+1]) != s);
    if (first) segStart[s] = e;
    if (last)  segEnd[s]   = e + 1;
}

// ---------------- stage 1: xt[hw][bc] = qw[h] * x[bc][hw] ----------------
// block = 128 lanes (bc), each block handles 16 consecutive hw -> each lane
// reads 16 consecutive floats (one cacheline), writes are 128-wide coalesced.
__global__ void k_prep(const float* __restrict__ x, const float* __restrict__ qw,
                       float* __restrict__ xt) {
    int bc  = threadIdx.x;            // 0..127
    int hw0 = blockIdx.x * 16;
    #pragma unroll
    for (int j = 0; j < 16; ++j) {
        int hw = hw0 + j;
        if (hw >= HWIN) return;
        int h = hw / WIN;
        xt[(size_t)hw * BC + bc] = qw[h] * x[(size_t)bc * HWIN + hw];
    }
}

// ---------------- stage 2: sparse gather into y[b][ck][hw] ----------------
// grid = (WOUT/2, NSEG); block = 256 = 128 bc lanes x 2 longitudes.
__global__ void k_gather(const float* __restrict__ xt, const float* __restrict__ vals,
                         const long long* __restrict__ idx,
                         const int* __restrict__ segStart, const int* __restrict__ segEnd,
                         int nnz, float* __restrict__ y) {
    int tid = threadIdx.x;
    int bc  = tid & (BC - 1);
    int pp  = tid >> 7;                        // 0..1
    int p   = blockIdx.x * 2 + pp;
    int s   = blockIdx.y;
    int k   = s / HOUT;
    int t   = s - k * HOUT;
    int e0 = segStart[s], e1 = segEnd[s];
    const long long* flat = idx + 2 * (size_t)nnz;
    float acc = 0.0f;
    for (int e = e0; e < e1; ++e) {
        int f    = (int)flat[e];
        int lon  = f % WIN;
        int base = f - lon;
        int col  = base + (lon + p * PSCALE) % WIN;
        acc = fmaf(vals[e], xt[(size_t)col * BC + bc], acc);
    }
    int b = bc >> 5, c = bc & 31;
    y[((size_t)(b * CK + c * KSIZE + k)) * NPAD + (size_t)t * WOUT + p] = acc;
}

// ---------------- stage 3: out[b][o][hw] = W[o][ck] x y[b][ck][hw] + bias ----
// One wave per 16x16 C tile; K=288 in steps of 4 via fp32 WMMA.
// Tiles: BATCH(4) x otile(2) x ntile(1024) = 8192 waves = 1024 blocks x 8 waves,
// exact coverage -> EXEC is all-ones at every WMMA.
__global__ void k_gemm(const float* __restrict__ y, const float* __restrict__ w,
                       const float* __restrict__ bias, float* __restrict__ out) {
    int wave = (blockIdx.x * blockDim.x + threadIdx.x) >> 5;
    int lane = threadIdx.x & 31;
    int nt = wave & (NPAD / 16 - 1);           // 0..1023
    int ot = (wave >> 10) & 1;                 // 0..1
    int b  = wave >> 11;                       // 0..3

    int hw0  = nt * 16;
    int half = lane >> 4;                      // lane block (0|1)
    int l    = lane & 15;

    v8f acc = {};
    // A: 16x4 fp32, M = lane&15, K = 2*half + vgpr   (ISA 7.12.2 layout)
    const float* wrow = w + (size_t)(ot * 16 + l) * CK + 2 * half;
    // B: 4x16 fp32, N = lane&15, K = 2*half + vgpr   (mirrored striping)
    const float* ycol = y + (size_t)b * CK * NPAD + hw0 + l;

    #pragma unroll 4
    for (int ck = 0; ck < CK; ck += 4) {
        v2f a, bb;
        a.x  = wrow[ck + 0];
        a.y  = wrow[ck + 1];
        bb.x = ycol[(size_t)(ck + 2 * half + 0) * NPAD];
        bb.y = ycol[(size_t)(ck + 2 * half + 1) * NPAD];
        acc = __builtin_amdgcn_wmma_f32_16x16x4_f32(
                  /*neg_a=*/false, a, /*neg_b=*/false, bb,
                  /*c_mod=*/(short)0, acc, /*reuse_a=*/false, /*reuse_b=*/false);
    }

    // D layout: M = vgpr + 8*half, N = lane&15
    int hw = hw0 + l;
    if (hw < HWOUT) {
        #pragma unroll
        for (int v = 0; v < 8; ++v) {
            int o = ot * 16 + v + 8 * half;
            out[((size_t)(b * COUT + o)) * HWOUT + hw] = acc[v] + bias[o];
        }
    }
}

extern "C" void kernel_launch(void* const* d_in, const int* in_sizes, int n_in,
                              void* d_out, int out_size, void* d_ws, size_t ws_size,
                              hipStream_t stream) {
    const float*     x    = (const float*)d_in[0];
    const float*     qw   = (const float*)d_in[1];   // (181,1)
    const float*     vals = (const float*)d_in[2];   // (NNZ,)
    const float*     wgt  = (const float*)d_in[3];   // (32,32,9) -> [o][ck]
    const float*     bias = (const float*)d_in[4];   // (32,)
    const long long* idx  = (const long long*)d_in[5]; // (3,NNZ) int64
    int nnz = in_sizes[2];
    float* out = (float*)d_out;

    // workspace layout
    float* xt = (float*)d_ws;                                // HWIN*BC floats
    float* y  = xt + (size_t)HWIN * BC;                      // BATCH*CK*NPAD floats
    int* segStart = (int*)(y + (size_t)BATCH * CK * NPAD);   // NSEG ints
    int* segEnd   = segStart + NSEG;

    int ny = BATCH * CK * NPAD;                              // 18,874,368
    k_zero    <<<2048, 256, 0, stream>>>(y, ny);
    k_seg_init<<<(NSEG + 255) / 256, 256, 0, stream>>>(segStart, segEnd);
    k_seg_mark<<<(nnz + 255) / 256, 256, 0, stream>>>(idx, nnz, segStart, segEnd);
    k_prep    <<<(HWIN + 15) / 16, 128, 0, stream>>>(x, qw, xt);
    k_gather  <<<dim3(WOUT / 2, NSEG), 256, 0, stream>>>(xt, vals, idx, segStart, segEnd, nnz, y);
    k_gemm    <<<(BATCH * 2 * (NPAD / 16)) / 8, 256, 0, stream>>>(y, wgt, bias, out);
}